// Solver_11158325035628
// MI455X (gfx1250) — compile-verified
//
#include <hip/hip_runtime.h>
#include <math.h>

// ---------------------------------------------------------------------------
// Dims
#define ZD 16
#define ND 466
#define HD 86
#define BSZ 2048
#define TT 32
#define NWG 16          // solver workgroups (co-resident, spin barrier)
#define WPB 8           // waves per solver block (256 threads)
#define KB_N 68         // 272 / 4 K-blocks for f(z) fused GEMM
#define ITERS 496       // 31 intervals * 16 trial steps

// Workspace layout (float offsets). Requires ~5.2 MB of d_ws.
static constexpr size_t OFF_ZS    = 0;                       // [32][2048][16]
static constexpr size_t OFF_HENC  = 1048576;                 // [2048][86]
static constexpr size_t OFF_WSW   = 1224704;                 // [68][32][2] swizzled W for solver
static constexpr size_t OFF_WD1SW = 1229056;                 // [6][4][32][2]
static constexpr size_t OFF_BD1P  = 1230592;                 // [96]
static constexpr size_t OFF_WD2SW = 1230688;                 // [30][24][32][2]
static constexpr size_t OFF_BD2P  = 1276768;                 // [480]
static constexpr size_t OFF_HB3   = 1277248;                 // [2048]
static constexpr size_t OFF_AMAT  = 1279296;                 // [256]
static constexpr size_t OFF_BOUT  = 1279552;                 // [4096]
static constexpr size_t OFF_PART  = 1283648;                 // [496][16] err partials
static constexpr size_t OFF_ARR   = 1291584;                 // [496] u32 arrival counters

// Dopri5 tableau (fp32)
static constexpr float A21f=(float)(1.0/5.0);
static constexpr float A31f=(float)(3.0/40.0),  A32f=(float)(9.0/40.0);
static constexpr float A41f=(float)(44.0/45.0), A42f=(float)(-56.0/15.0), A43f=(float)(32.0/9.0);
static constexpr float A51f=(float)(19372.0/6561.0), A52f=(float)(-25360.0/2187.0),
                       A53f=(float)(64448.0/6561.0), A54f=(float)(-212.0/729.0);
static constexpr float A61f=(float)(9017.0/3168.0), A62f=(float)(-355.0/33.0),
                       A63f=(float)(46732.0/5247.0), A64f=(float)(49.0/176.0),
                       A65f=(float)(-5103.0/18656.0);
static constexpr float B1f=(float)(35.0/384.0), B3f=(float)(500.0/1113.0), B4f=(float)(125.0/192.0),
                       B5f=(float)(-2187.0/6784.0), B6f=(float)(11.0/84.0);
static constexpr float E1f=(float)(71.0/57600.0), E3f=(float)(-71.0/16695.0), E4f=(float)(71.0/1920.0),
                       E5f=(float)(-17253.0/339200.0), E6f=(float)(22.0/525.0), E7f=(float)(-1.0/40.0);

typedef __attribute__((ext_vector_type(2))) float    v2f;
typedef __attribute__((ext_vector_type(8))) float    v8f;
typedef __attribute__((ext_vector_type(4))) unsigned v4u;
typedef __attribute__((ext_vector_type(4))) int      v4i;
typedef __attribute__((ext_vector_type(8))) int      v8i;

__device__ __forceinline__ float leaky(float x) { return x >= 0.f ? x : 0.2f * x; }

// D = A(16x4) * B(4x16) + C, fp32 WMMA.
__device__ __forceinline__ v8f wmma4(v2f a, v2f b, v8f c) {
  return __builtin_amdgcn_wmma_f32_16x16x4_f32(false, a, false, b, (short)0, c, false, false);
}

// ---------------------------------------------------------------------------
// Encoder stage 1: h = leaky(n0 @ We1 + be1)   (one block per batch row)
__global__ void k_enc1(const float* __restrict__ n0, const float* __restrict__ We1,
                       const float* __restrict__ be1, float* __restrict__ ws) {
  int b = blockIdx.x, hc = threadIdx.x;
  if (hc >= HD) return;
  const float* row = n0 + (size_t)b * ND;
  float acc = be1[hc];
  for (int n = 0; n < ND; ++n) acc = fmaf(row[n], We1[n * HD + hc], acc);
  ws[OFF_HENC + (size_t)b * HD + hc] = leaky(acc);
}

// Encoder stage 2: z0 = h @ We2 + be2  -> zs[0]
__global__ void k_enc2(const float* __restrict__ We2, const float* __restrict__ be2,
                       float* __restrict__ ws) {
  int tid = blockIdx.x * 256 + threadIdx.x;
  if (tid >= BSZ * ZD) return;
  int b = tid >> 4, zc = tid & 15;
  const float* h = ws + OFF_HENC + (size_t)b * HD;
  float acc = be2[zc];
  for (int c = 0; c < HD; ++c) acc = fmaf(h[c], We2[c * ZD + zc], acc);
  ws[OFF_ZS + (size_t)b * ZD + zc] = acc;   // no leaky on z0
}

// ---------------------------------------------------------------------------
// A-net / B-net front (p is 4 values) -> Amat[256] and hb3[2048]
__global__ void k_prep_small(const float* __restrict__ p,
                             const float* __restrict__ Wa1, const float* __restrict__ ba1,
                             const float* __restrict__ Wa2, const float* __restrict__ ba2,
                             const float* __restrict__ Wa3, const float* __restrict__ ba3,
                             const float* __restrict__ Wb1, const float* __restrict__ bb1,
                             const float* __restrict__ Wb2, const float* __restrict__ bb2,
                             const float* __restrict__ Wb3, const float* __restrict__ bb3,
                             float* __restrict__ ws) {
  __shared__ float ha1[16], ha2[128], hb1[16], hb2[64];
  int t = threadIdx.x;
  if (t < 16) {
    float a = ba1[t]; for (int k = 0; k < 4; ++k) a = fmaf(p[k], Wa1[k * 16 + t], a);
    ha1[t] = leaky(a);
    float b = bb1[t]; for (int k = 0; k < 4; ++k) b = fmaf(p[k], Wb1[k * 16 + t], b);
    hb1[t] = leaky(b);
  }
  __syncthreads();
  if (t < 128) {
    float a = ba2[t]; for (int k = 0; k < 16; ++k) a = fmaf(ha1[k], Wa2[k * 128 + t], a);
    ha2[t] = leaky(a);
  }
  if (t < 64) {
    float b = bb2[t]; for (int k = 0; k < 16; ++k) b = fmaf(hb1[k], Wb2[k * 64 + t], b);
    hb2[t] = leaky(b);
  }
  __syncthreads();
  { // Amat = leaky(ha2 @ Wa3 + ba3), 256 outputs
    float a = ba3[t];
    for (int k = 0; k < 128; ++k) a = fmaf(ha2[k], Wa3[k * 256 + t], a);
    ws[OFF_AMAT + t] = leaky(a);
  }
  for (int o = t; o < 2048; o += 256) {
    float b = bb3[o];
    for (int k = 0; k < 64; ++k) b = fmaf(hb2[k], Wb3[k * 2048 + o], b);
    ws[OFF_HB3 + o] = leaky(b);
  }
}

// Bout = leaky(hb3 @ Wb4 + bb4)  [4096]
__global__ void k_prep_B(const float* __restrict__ Wb4, const float* __restrict__ bb4,
                         float* __restrict__ ws) {
  int o = blockIdx.x * 256 + threadIdx.x;
  const float* hb3 = ws + OFF_HB3;
  float acc = bb4[o];
  for (int k = 0; k < 2048; ++k) acc = fmaf(hb3[k], Wb4[(size_t)k * 4096 + o], acc);
  ws[OFF_BOUT + o] = leaky(acc);
}

// ---------------------------------------------------------------------------
// Pack W matrices into WMMA B-operand lane layout:
// reg v of lane L holds K = v + 2*(L/16), N = L%16  -> stored as [kb][lane][2].
__global__ void k_pack(const float* __restrict__ Wd1, const float* __restrict__ bd1,
                       const float* __restrict__ Wd2, const float* __restrict__ bd2,
                       float* __restrict__ ws) {
  int tid = blockIdx.x * 256 + threadIdx.x;
  const float* Amat = ws + OFF_AMAT;
  const float* Bout = ws + OFF_BOUT;
  if (tid < KB_N * 32) {                       // solver W: rows 0..15 = A^T, 16+j*16+k = B[i][j][k]
    int kb = tid >> 5, L = tid & 31;
    int hh = L >> 4, n = L & 15;
    int kk = kb * 4 + 2 * hh;
    float v0 = (kk     < 16) ? Amat[n * 16 + kk]     : Bout[n * 256 + (kk - 16)];
    float v1 = (kk + 1 < 16) ? Amat[n * 16 + kk + 1] : Bout[n * 256 + (kk + 1 - 16)];
    ws[OFF_WSW + (size_t)tid * 2]     = v0;
    ws[OFF_WSW + (size_t)tid * 2 + 1] = v1;
  } else if (tid < 2944) {                     // Wd1 swizzled, N padded 86->96
    int q = tid - 2176;
    int L = q & 31, kb = (q >> 5) & 3, nt = q >> 7;
    int hh = L >> 4, n = nt * 16 + (L & 15);
    int kk = kb * 4 + 2 * hh;
    ws[OFF_WD1SW + (size_t)q * 2]     = (n < HD) ? Wd1[kk * HD + n] : 0.f;
    ws[OFF_WD1SW + (size_t)q * 2 + 1] = (n < HD) ? Wd1[(kk + 1) * HD + n] : 0.f;
  } else if (tid < 3040) {                     // bd1 padded to 96
    int c0 = tid - 2944;
    ws[OFF_BD1P + c0] = (c0 < HD) ? bd1[c0] : 0.f;
  } else if (tid < 3520) {                     // bd2 padded to 480
    int c0 = tid - 3040;
    ws[OFF_BD2P + c0] = (c0 < ND) ? bd2[c0] : 0.f;
  } else if (tid < 26560) {                    // Wd2 swizzled, K 86->96, N 466->480
    int q = tid - 3520;
    int L = q & 31, kb = (q >> 5) % 24, nt = q / 768;
    int hh = L >> 4, n = nt * 16 + (L & 15);
    int kk = kb * 4 + 2 * hh;
    ws[OFF_WD2SW + (size_t)q * 2]     = (kk     < HD && n < ND) ? Wd2[(size_t)kk * ND + n] : 0.f;
    ws[OFF_WD2SW + (size_t)q * 2 + 1] = (kk + 1 < HD && n < ND) ? Wd2[(size_t)(kk + 1) * ND + n] : 0.f;
  }
}

// Zero the per-iteration partials / arrival counters (must run before solver each call).
__global__ void k_init(float* __restrict__ ws) {
  int tid = blockIdx.x * 256 + threadIdx.x;
  if (tid < ITERS * NWG)            ws[OFF_PART + tid] = 0.f;
  else if (tid < ITERS * NWG + ITERS)
    ((unsigned*)(ws + OFF_ARR))[tid - ITERS * NWG] = 0u;
}

// ---------------------------------------------------------------------------
// f(z) = z A^T + einsum(B,z,z) for one 16-row tile, fused as G[16,272] @ W[272,16].
// y is in WMMA C/D layout. Transpose through this wave's private LDS slot.
// Branch-free: 4 linear K-blocks, then 16 j-groups of 4 quadratic K-blocks that
// share one z_j scalar and the 4 hoisted z_k pairs.
__device__ __forceinline__ v8f f_eval(v8f y, float* st, const float* wsw, int lane) {
  const int h = lane >> 4, c = lane & 15;
  asm volatile("" ::: "memory");
  __builtin_amdgcn_wave_barrier();
#pragma unroll
  for (int v = 0; v < 8; ++v) st[(v + 8 * h) * 16 + c] = y[v];
  asm volatile("s_wait_dscnt 0" ::: "memory");
  __builtin_amdgcn_wave_barrier();
  const float* row  = st + c * 16;           // this lane supplies A rows for M = lane%16
  const float* rowh = row + 2 * h;           // K-pair base for this lane half
  const v2f*   wb   = (const v2f*)wsw + lane;

  // z_k pairs for this half: elements {2h,2h+1}+{0,4,8,12} (used by both parts)
  v2f zp0 = *(const v2f*)(rowh + 0);
  v2f zp1 = *(const v2f*)(rowh + 4);
  v2f zp2 = *(const v2f*)(rowh + 8);
  v2f zp3 = *(const v2f*)(rowh + 12);

  v8f acc = {0.f, 0.f, 0.f, 0.f, 0.f, 0.f, 0.f, 0.f};
  // linear part: G[kk] = z[kk], kb = 0..3
  acc = wmma4(zp0, wb[0 * 32], acc);
  acc = wmma4(zp1, wb[1 * 32], acc);
  acc = wmma4(zp2, wb[2 * 32], acc);
  acc = wmma4(zp3, wb[3 * 32], acc);
  // quadratic part: kb = 4 + j*4 + m, G = z_j * z_{m*4+2h (+1)}
  const v2f* wq = wb + 4 * 32;
#pragma unroll 1
  for (int j = 0; j < 16; ++j) {
    float zj = row[j];
    v2f a;
    a[0] = zj * zp0[0]; a[1] = zj * zp0[1];
    acc = wmma4(a, wq[(j * 4 + 0) * 32], acc);
    a[0] = zj * zp1[0]; a[1] = zj * zp1[1];
    acc = wmma4(a, wq[(j * 4 + 1) * 32], acc);
    a[0] = zj * zp2[0]; a[1] = zj * zp2[1];
    acc = wmma4(a, wq[(j * 4 + 2) * 32], acc);
    a[0] = zj * zp3[0]; a[1] = zj * zp3[1];
    acc = wmma4(a, wq[(j * 4 + 3) * 32], acc);
  }
  return acc;
}

__global__ __launch_bounds__(256) void k_solver(const float* __restrict__ tstep,
                                                float* __restrict__ ws) {
  __shared__ float s_wsw[KB_N * 32 * 2];
  __shared__ float s_t[WPB][256];
  __shared__ float s_red[WPB];
  __shared__ float s_en;

  const int tidx = threadIdx.x;
  const int lane = tidx & 31, wave = tidx >> 5;
  const int wg = blockIdx.x;
  const int r0 = (wg * WPB + wave) * 16;
  const int h = lane >> 4, c = lane & 15;

  // Stage swizzled W (17408 B) into LDS via the Tensor Data Mover (wave 0 only;
  // TDM ignores EXEC, TENSORcnt is per-wave).
#if __has_builtin(__builtin_amdgcn_tensor_load_to_lds)
  if (wave == 0) {
    unsigned lds_off;
    {
      __attribute__((address_space(3))) float* p3 =
          (__attribute__((address_space(3))) float*)s_wsw;
      lds_off = (unsigned)(unsigned long long)p3;
    }
    unsigned long long ga = (unsigned long long)(const void*)(ws + OFF_WSW);
    v4u g0;
    g0[0] = 1u;                                   // count=1, user descriptor
    g0[1] = lds_off;                              // lds_addr
    g0[2] = (unsigned)(ga & 0xffffffffull);       // global_addr[31:0]
    g0[3] = (unsigned)((ga >> 32) & 0x1ffffffull) // global_addr[56:32]
            | (2u << 30);                         // type = 2 ("image")
    // data_size=4B; tensor_dim0=tile_dim0=4352 elems; dim1=1; strides=4352
    v8i g1;
    g1[0] = 0x20000;     // data_size=2 (4B)
    g1[1] = 0x11000000;  // tensor_dim0[15:0]=0x1100 in bits[31:16]
    g1[2] = 0x00010000;  // tensor_dim0[31:16]=0, tensor_dim1[15:0]=1
    g1[3] = 0x11000000;  // tensor_dim1[31:16]=0, tile_dim0=0x1100
    g1[4] = 1;           // tile_dim1=1, tile_dim2=0
    g1[5] = 4352;        // tensor_dim0_stride[31:0]
    g1[6] = 0x11000000;  // stride0 hi=0, tensor_dim1_stride lo16=0x1100
    g1[7] = 0;           // tensor_dim1_stride hi
    v4i gz; gz[0] = 0; gz[1] = 0; gz[2] = 0; gz[3] = 0;
#if defined(__clang_major__) && (__clang_major__ >= 23)
    v8i gz8; for (int i = 0; i < 8; ++i) gz8[i] = 0;
    __builtin_amdgcn_tensor_load_to_lds(g0, g1, gz, gz, gz8, 0);
#else
    __builtin_amdgcn_tensor_load_to_lds(g0, g1, gz, gz, 0);
#endif
    __builtin_amdgcn_s_wait_tensorcnt(0);
  }
#else
  for (int i = tidx; i < KB_N * 32 * 2; i += 256) s_wsw[i] = ws[OFF_WSW + i];
#endif
  __syncthreads();

  float* zs = ws + OFF_ZS;
  float* part = ws + OFF_PART;
  unsigned* arrv = (unsigned*)(ws + OFF_ARR);

  v8f y;
#pragma unroll
  for (int v = 0; v < 8; ++v) y[v] = zs[(size_t)(r0 + v + 8 * h) * 16 + c];

  float dt = (tstep[1] - tstep[0]) * 0.3f;
  int it = 0;

  for (int iv = 0; iv < TT - 1; ++iv) {
    float t = tstep[iv];
    const float t1 = tstep[iv + 1];
    for (int inner = 0; inner < 16; ++inner, ++it) {
      const bool done = (t >= t1 - 1e-10f);
      const float dtc = done ? 0.f : fminf(dt, t1 - t);

      v8f k1 = f_eval(y, s_t[wave], s_wsw, lane);
      v8f yt;
#pragma unroll
      for (int v = 0; v < 8; ++v) yt[v] = y[v] + dtc * (A21f * k1[v]);
      v8f k2 = f_eval(yt, s_t[wave], s_wsw, lane);
#pragma unroll
      for (int v = 0; v < 8; ++v) yt[v] = y[v] + dtc * (A31f * k1[v] + A32f * k2[v]);
      v8f k3 = f_eval(yt, s_t[wave], s_wsw, lane);
#pragma unroll
      for (int v = 0; v < 8; ++v)
        yt[v] = y[v] + dtc * (A41f * k1[v] + A42f * k2[v] + A43f * k3[v]);
      v8f k4 = f_eval(yt, s_t[wave], s_wsw, lane);
#pragma unroll
      for (int v = 0; v < 8; ++v)
        yt[v] = y[v] + dtc * (A51f * k1[v] + A52f * k2[v] + A53f * k3[v] + A54f * k4[v]);
      v8f k5 = f_eval(yt, s_t[wave], s_wsw, lane);
#pragma unroll
      for (int v = 0; v < 8; ++v)
        yt[v] = y[v] + dtc * (A61f * k1[v] + A62f * k2[v] + A63f * k3[v] + A64f * k4[v] + A65f * k5[v]);
      v8f k6 = f_eval(yt, s_t[wave], s_wsw, lane);
      v8f y5;
#pragma unroll
      for (int v = 0; v < 8; ++v)
        y5[v] = y[v] + dtc * (B1f * k1[v] + B3f * k3[v] + B4f * k4[v] + B5f * k5[v] + B6f * k6[v]);
      v8f k7 = f_eval(y5, s_t[wave], s_wsw, lane);

      float local = 0.f;
#pragma unroll
      for (int v = 0; v < 8; ++v) {
        float err = dtc * (E1f * k1[v] + E3f * k3[v] + E4f * k4[v] + E5f * k5[v] +
                           E6f * k6[v] + E7f * k7[v]);
        float tol = 1e-20f + 1e-5f * fmaxf(fabsf(y[v]), fabsf(y5[v]));
        float r = err / tol;
        local += r * r;
      }
      // wave reduce (wave32)
      for (int off = 16; off > 0; off >>= 1) local += __shfl_xor(local, off);
      if (lane == 0) s_red[wave] = local;
      __syncthreads();
      if (tidx == 0) {
        float s = 0.f;
        for (int w = 0; w < WPB; ++w) s += s_red[w];
        __hip_atomic_store(&part[it * NWG + wg], s, __ATOMIC_RELAXED, __HIP_MEMORY_SCOPE_AGENT);
        __hip_atomic_fetch_add(&arrv[it], 1u, __ATOMIC_ACQ_REL, __HIP_MEMORY_SCOPE_AGENT);
        while (__hip_atomic_load(&arrv[it], __ATOMIC_ACQUIRE, __HIP_MEMORY_SCOPE_AGENT) < NWG)
          __builtin_amdgcn_s_sleep(1);
        float tot = 0.f;                      // fixed-order sum -> deterministic
        for (int g = 0; g < NWG; ++g)
          tot += __hip_atomic_load(&part[it * NWG + g], __ATOMIC_RELAXED, __HIP_MEMORY_SCOPE_AGENT);
        s_en = sqrtf(tot * (1.0f / (float)(BSZ * ZD)));
      }
      __syncthreads();
      const float en = s_en;
      const bool accept = (en <= 1.0f) && !done;
      if (accept) {
        t += dtc;
#pragma unroll
        for (int v = 0; v < 8; ++v) y[v] = y5[v];
      }
      float fac = fminf(fmaxf(0.9f * powf(fmaxf(en, 1e-12f), -0.2f), 0.2f), 10.0f);
      if (!done) dt *= fac;
    }
#pragma unroll
    for (int v = 0; v < 8; ++v)
      zs[((size_t)(iv + 1) * BSZ + r0 + v + 8 * h) * 16 + c] = y[v];
  }
}

// ---------------------------------------------------------------------------
// Decoder: out = leaky(zs @ Wd1 + bd1) @ Wd2 + bd2 ; one 16-row M-tile per wave.
__global__ __launch_bounds__(256) void k_decoder(const float* __restrict__ ws,
                                                 float* __restrict__ out) {
  __shared__ float s_wd1[6 * 4 * 32 * 2];
  __shared__ float s_bd1[96];
  __shared__ float s_bd2[480];
  __shared__ float s_h[WPB][16 * 96];

  const int tidx = threadIdx.x;
  const int lane = tidx & 31, wave = tidx >> 5;
  const int h = lane >> 4, c = lane & 15;

  for (int i = tidx; i < 6 * 4 * 32 * 2; i += 256) s_wd1[i] = ws[OFF_WD1SW + i];
  if (tidx < 96)  s_bd1[tidx] = ws[OFF_BD1P + tidx];
  for (int i = tidx; i < 480; i += 256) s_bd2[i] = ws[OFF_BD2P + i];
  __syncthreads();

  const int r0 = (blockIdx.x * WPB + wave) * 16;
  const float* zs = ws + OFF_ZS;

  float zr[16];
  {
    const float* zp = zs + (size_t)(r0 + c) * 16;
#pragma unroll
    for (int j = 0; j < 16; ++j) zr[j] = zp[j];
  }

  // GEMM1: h_tile[16x96] = leaky(z[16x16] @ Wd1p[16x96] + bd1p)
  float* hp = &s_h[wave][0];
#pragma unroll
  for (int nt = 0; nt < 6; ++nt) {
    v8f acc = {0.f, 0.f, 0.f, 0.f, 0.f, 0.f, 0.f, 0.f};
#pragma unroll
    for (int kb = 0; kb < 4; ++kb) {
      v2f a;
      a[0] = h ? zr[kb * 4 + 2] : zr[kb * 4];
      a[1] = h ? zr[kb * 4 + 3] : zr[kb * 4 + 1];
      v2f b = *(const v2f*)(s_wd1 + ((nt * 4 + kb) * 32 + lane) * 2);
      acc = wmma4(a, b, acc);
    }
    const int col = nt * 16 + c;
#pragma unroll
    for (int v = 0; v < 8; ++v)
      hp[(v + 8 * h) * 96 + col] = leaky(acc[v] + s_bd1[col]);
  }
  asm volatile("s_wait_dscnt 0" ::: "memory");
  __builtin_amdgcn_wave_barrier();

  // GEMM2: out_tile = h_tile[16x96] @ Wd2p[96x480] + bd2p
  const float* hrow = hp + c * 96;
  const v2f* wd2 = (const v2f*)(ws + OFF_WD2SW);
  for (int nt2 = 0; nt2 < 30; ++nt2) {
    if (nt2 + 1 < 30)  // prefetch next N-tile's swizzled Wd2 block into cache
      __builtin_prefetch((const void*)(wd2 + (size_t)(nt2 + 1) * 24 * 32), 0, 0);
    v8f acc = {0.f, 0.f, 0.f, 0.f, 0.f, 0.f, 0.f, 0.f};
#pragma unroll 4
    for (int kb = 0; kb < 24; ++kb) {
      v2f a = *(const v2f*)(hrow + kb * 4 + 2 * h);
      v2f b = wd2[(size_t)(nt2 * 24 + kb) * 32 + lane];
      acc = wmma4(a, b, acc);
    }
    const int col = nt2 * 16 + c;
    if (col < ND) {
#pragma unroll
      for (int v = 0; v < 8; ++v)
        out[(size_t)(r0 + v + 8 * h) * ND + col] = acc[v] + s_bd2[col];
    }
  }
}

// ---------------------------------------------------------------------------
extern "C" void kernel_launch(void* const* d_in, const int* in_sizes, int n_in,
                              void* d_out, int out_size, void* d_ws, size_t ws_size,
                              hipStream_t stream) {
  (void)in_sizes; (void)n_in; (void)out_size; (void)ws_size;
  const float* n0  = (const float*)d_in[0];
  const float* p   = (const float*)d_in[1];
  const float* ts  = (const float*)d_in[2];
  const float* We1 = (const float*)d_in[3];  const float* be1 = (const float*)d_in[4];
  const float* We2 = (const float*)d_in[5];  const float* be2 = (const float*)d_in[6];
  const float* Wd1 = (const float*)d_in[7];  const float* bd1 = (const float*)d_in[8];
  const float* Wd2 = (const float*)d_in[9];  const float* bd2 = (const float*)d_in[10];
  const float* Wa1 = (const float*)d_in[11]; const float* ba1 = (const float*)d_in[12];
  const float* Wa2 = (const float*)d_in[13]; const float* ba2 = (const float*)d_in[14];
  const float* Wa3 = (const float*)d_in[15]; const float* ba3 = (const float*)d_in[16];
  const float* Wb1 = (const float*)d_in[17]; const float* bb1 = (const float*)d_in[18];
  const float* Wb2 = (const float*)d_in[19]; const float* bb2 = (const float*)d_in[20];
  const float* Wb3 = (const float*)d_in[21]; const float* bb3 = (const float*)d_in[22];
  const float* Wb4 = (const float*)d_in[23]; const float* bb4 = (const float*)d_in[24];
  float* out = (float*)d_out;
  float* ws  = (float*)d_ws;

  k_enc1<<<BSZ, 96, 0, stream>>>(n0, We1, be1, ws);
  k_enc2<<<(BSZ * ZD + 255) / 256, 256, 0, stream>>>(We2, be2, ws);
  k_prep_small<<<1, 256, 0, stream>>>(p, Wa1, ba1, Wa2, ba2, Wa3, ba3,
                                      Wb1, bb1, Wb2, bb2, Wb3, bb3, ws);
  k_prep_B<<<16, 256, 0, stream>>>(Wb4, bb4, ws);
  k_pack<<<104, 256, 0, stream>>>(Wd1, bd1, Wd2, bd2, ws);
  k_init<<<33, 256, 0, stream>>>(ws);
  k_solver<<<NWG, 256, 0, stream>>>(ts, ws);
  k_decoder<<<(TT * BSZ / 16) / WPB, 256, 0, stream>>>(ws, out);
}